// GCN_61572651155681
// MI455X (gfx1250) — compile-verified
//
#include <hip/hip_runtime.h>
#include <hip/hip_bf16.h>

// ---------------------------------------------------------------------------
// GCN forward on MI455X (gfx1250, wave32).
//   logits, embeddings = GCN(x, edge_index, W1,b1, W2,b2, Wc,bc)
// GEMMs use native fp32 WMMA (V_WMMA_F32_16X16X4_F32), one wave per 16-row
// strip producing the full 16x128 output block (8 accumulators, A read once).
// B is pre-packed into per-lane fragment order so each B fragment is one
// coalesced b64 load; inner loop = 9 clauseable b64 loads + 8 WMMAs.
// Edge aggregation: one wave per edge (lane = float4 feature slice) with
// f32 global atomics; the 51.2MB feature matrices live in the 192MB L2.
// ---------------------------------------------------------------------------

#define N_NODES   100000
#define N_EDGES   1600000
#define HIDDEN    128

typedef __attribute__((ext_vector_type(2))) float v2f;
typedef __attribute__((ext_vector_type(8))) float v8f;

// ---------------------------- small utility kernels ------------------------

__global__ void gcn_fill_f32(float* __restrict__ p, float v, int n) {
    int i = blockIdx.x * blockDim.x + threadIdx.x;
    if (i < n) p[i] = v;
}

__global__ void gcn_zero_f32(float* __restrict__ p, int n) {
    int i = blockIdx.x * blockDim.x + threadIdx.x;
    if (i < n) p[i] = 0.0f;
}

// deg[dst] += 1 per edge (dinv preinitialized to 1.0 for the self-loop)
__global__ void gcn_degree(const long long* __restrict__ dst, float* __restrict__ deg, int E) {
    int e = blockIdx.x * blockDim.x + threadIdx.x;
    if (e < E) atomicAdd(&deg[(int)dst[e]], 1.0f);
}

__global__ void gcn_rsqrt_inplace(float* __restrict__ d, int n) {
    int i = blockIdx.x * blockDim.x + threadIdx.x;
    if (i < n) d[i] = rsqrtf(d[i]);   // deg >= 1 always (self-loop)
}

// --------------------------- B fragment pre-pack ---------------------------
// Repack B[K x 128] (row-major) into WMMA B-fragment order:
//   Bpk[((kstep*8 + t)*32 + lane)*2 + j] = B[kstep*4 + 2*hi + j][t*16 + n]
// with hi = lane>>4, n = lane&15. Each wave's fragment load is then one
// contiguous 256B chunk (b64 per lane).
__global__ void gcn_pack_B(const float* __restrict__ B, float* __restrict__ Bpk, int K) {
    int idx = blockIdx.x * blockDim.x + threadIdx.x;   // K*128 elements
    if (idx >= K * HIDDEN) return;
    const int kstep = idx >> 9;          // / 512
    const int r     = idx & 511;
    const int t     = r >> 6;            // / 64
    const int r2    = r & 63;
    const int lane  = r2 >> 1;
    const int j     = r2 & 1;
    const int hi    = lane >> 4;
    const int n     = lane & 15;
    Bpk[idx] = B[(size_t)(kstep * 4 + 2 * hi + j) * HIDDEN + t * 16 + n];
}

// ------------------------------ fp32 WMMA GEMM -----------------------------
// C[M x 128] = A[M x K] * Bpk (pre-packed), row-major C, no bias.
// One wave per 16-row strip of C; 8 v8f accumulators (all column tiles).
//
// A-frag (16x4 f32): lane m = lane&15, hi = lane>>4; VGPR v = A[m][k+2*hi+v].
// C/D (16x16 f32, 8 VGPRs): VGPR v -> row 8*hi + v, col = lane&15.
__global__ void gcn_gemm_wmma_f32(const float* __restrict__ A,
                                  const float* __restrict__ Bpk,
                                  float* __restrict__ C,
                                  int M, int K) {
    const int wave = (blockIdx.x * blockDim.x + threadIdx.x) >> 5;
    const int lane = threadIdx.x & 31;
    if (wave * 16 >= M) return;              // whole-wave uniform: EXEC all-1s

    const int n  = lane & 15;                // also the A row-in-tile (m)
    const int hi = lane >> 4;                // 0 or 1

    const float* Arow = A + (size_t)(wave * 16 + n) * K;
    const v2f*   Bfr  = (const v2f*)Bpk;

    v8f acc[8] = {};
    for (int kstep = 0; kstep < K / 4; ++kstep) {
        v2f a;
        a.x = Arow[kstep * 4 + 2 * hi + 0];
        a.y = Arow[kstep * 4 + 2 * hi + 1];

        v2f bfrag[8];
#pragma unroll
        for (int t = 0; t < 8; ++t)
            bfrag[t] = Bfr[(size_t)(kstep * 8 + t) * 32 + lane];

#pragma unroll
        for (int t = 0; t < 8; ++t)
            acc[t] = __builtin_amdgcn_wmma_f32_16x16x4_f32(
                /*neg_a=*/false, a, /*neg_b=*/false, bfrag[t],
                /*c_mod=*/(short)0, acc[t], /*reuse_a=*/false, /*reuse_b=*/false);
    }

    float* Crow = C + (size_t)(wave * 16) * HIDDEN;
    const int rbase = hi * 8;
#pragma unroll
    for (int t = 0; t < 8; ++t)
#pragma unroll
        for (int v = 0; v < 8; ++v)
            Crow[(size_t)(rbase + v) * HIDDEN + t * 16 + n] = acc[t][v];
}

// --------------------------- edge scatter-aggregate ------------------------
// One wave per edge: lane l owns features [4l, 4l+4).
//   out[dst] += dinv[src]*dinv[dst] * h[src]   (f32 atomics into L2)
__global__ void gcn_edge_aggregate(const long long* __restrict__ src,
                                   const long long* __restrict__ dst,
                                   const float* __restrict__ dinv,
                                   const float* __restrict__ h,
                                   float* __restrict__ out, int E) {
    const int e    = (blockIdx.x * blockDim.x + threadIdx.x) >> 5;
    const int lane = threadIdx.x & 31;
    if (e >= E) return;
    const int s = (int)src[e];
    const int d = (int)dst[e];
    const float nrm = dinv[s] * dinv[d];
    const float4 v = ((const float4*)(h + (size_t)s * HIDDEN))[lane];
    float* op = out + (size_t)d * HIDDEN + lane * 4;
    atomicAdd(op + 0, nrm * v.x);
    atomicAdd(op + 1, nrm * v.y);
    atomicAdd(op + 2, nrm * v.z);
    atomicAdd(op + 3, nrm * v.w);
}

// self-loop term + bias + ReLU:  out = relu(out + dinv[i]^2 * h[i] + bias)
__global__ void gcn_epilogue(float* __restrict__ out,
                             const float* __restrict__ h,
                             const float* __restrict__ dinv,
                             const float* __restrict__ bias, int N) {
    const int idx  = blockIdx.x * blockDim.x + threadIdx.x;   // N*32 threads
    const int node = idx >> 5;
    const int f4   = (idx & 31) * 4;
    if (node >= N) return;
    const float di = dinv[node];
    const float w  = di * di;
    const size_t base = (size_t)node * HIDDEN + f4;
#pragma unroll
    for (int j = 0; j < 4; ++j) {
        float v = out[base + j] + w * h[base + j] + bias[f4 + j];
        out[base + j] = v > 0.0f ? v : 0.0f;
    }
}

// logits[i][c] = dot(emb[i], Wc[:,c]) + bc[c]    (Wc is [128,2] row-major)
__global__ void gcn_classifier(const float* __restrict__ emb,
                               const float* __restrict__ Wc,
                               const float* __restrict__ bc,
                               float* __restrict__ logits, int N) {
    const int idx  = blockIdx.x * blockDim.x + threadIdx.x;
    const int node = idx >> 1;
    const int c    = idx & 1;
    if (node >= N) return;
    const float4* e4 = (const float4*)(emb + (size_t)node * HIDDEN);
    float acc = bc[c];
#pragma unroll 8
    for (int f = 0; f < HIDDEN / 4; ++f) {
        const float4 v = e4[f];
        acc += v.x * Wc[(4 * f + 0) * 2 + c];
        acc += v.y * Wc[(4 * f + 1) * 2 + c];
        acc += v.z * Wc[(4 * f + 2) * 2 + c];
        acc += v.w * Wc[(4 * f + 3) * 2 + c];
    }
    logits[(size_t)node * 2 + c] = acc;
}

// --------------------------------- driver ----------------------------------

extern "C" void kernel_launch(void* const* d_in, const int* in_sizes, int n_in,
                              void* d_out, int out_size, void* d_ws, size_t ws_size,
                              hipStream_t stream) {
    const float*     x  = (const float*)d_in[0];
    const long long* ei = (const long long*)d_in[1];   // [2, E] int64 per reference
    const float*     W1 = (const float*)d_in[2];
    const float*     b1 = (const float*)d_in[3];
    const float*     W2 = (const float*)d_in[4];
    const float*     b2 = (const float*)d_in[5];
    const float*     Wc = (const float*)d_in[6];
    const float*     bc = (const float*)d_in[7];

    const int N = N_NODES, E = N_EDGES;
    const long long* esrc = ei;       // row 0
    const long long* edst = ei + E;   // row 1

    float* logits = (float*)d_out;                     // [N,2]
    float* emb    = (float*)d_out + (size_t)N * 2;     // [N,128]

    // workspace: dinv+Bpk (1MB region) | bufA (51.2MB) | bufB (51.2MB)
    char*  ws   = (char*)d_ws;
    float* dinv = (float*)ws;                               // N floats (400KB)
    float* Bpk  = (float*)(ws + 512 * 1024);                // up to 64KB packed B
    float* bufA = (float*)(ws + (1u << 20));
    float* bufB = (float*)(ws + (1u << 20) + (size_t)N * HIDDEN * sizeof(float));

    const int NH    = N * HIDDEN;      // 12.8M elements
    const int waves = N / 16;          // 6250 strip-waves per GEMM

    // 1) symmetric degree normalization
    gcn_fill_f32<<<(N + 255) / 256, 256, 0, stream>>>(dinv, 1.0f, N);
    gcn_degree<<<(E + 255) / 256, 256, 0, stream>>>(edst, dinv, E);
    gcn_rsqrt_inplace<<<(N + 255) / 256, 256, 0, stream>>>(dinv, N);

    // 2) layer 1: h1 = x @ W1  (WMMA, K=32), aggregate, +b1, ReLU -> bufB
    gcn_pack_B<<<(32 * HIDDEN + 255) / 256, 256, 0, stream>>>(W1, Bpk, 32);
    gcn_gemm_wmma_f32<<<(waves * 32 + 255) / 256, 256, 0, stream>>>(x, Bpk, bufA, N, 32);
    gcn_zero_f32<<<(NH + 255) / 256, 256, 0, stream>>>(bufB, NH);
    gcn_edge_aggregate<<<(E * 32 + 255) / 256, 256, 0, stream>>>(esrc, edst, dinv, bufA, bufB, E);
    gcn_epilogue<<<(N * 32 + 255) / 256, 256, 0, stream>>>(bufB, bufA, dinv, b1, N);

    // 3) layer 2: h2 = h1 @ W2 (WMMA, K=128), aggregate, +b2, ReLU -> embeddings
    gcn_pack_B<<<(128 * HIDDEN + 255) / 256, 256, 0, stream>>>(W2, Bpk, 128);
    gcn_gemm_wmma_f32<<<(waves * 32 + 255) / 256, 256, 0, stream>>>(bufB, Bpk, bufA, N, 128);
    gcn_zero_f32<<<(NH + 255) / 256, 256, 0, stream>>>(emb, NH);
    gcn_edge_aggregate<<<(E * 32 + 255) / 256, 256, 0, stream>>>(esrc, edst, dinv, bufA, emb, E);
    gcn_epilogue<<<(N * 32 + 255) / 256, 256, 0, stream>>>(emb, bufA, dinv, b2, N);

    // 4) classifier: logits = emb @ Wc + bc
    gcn_classifier<<<(2 * N + 255) / 256, 256, 0, stream>>>(emb, Wc, bc, logits, N);
}